// InverseInDegreeEdgeWeighting_89386859364991
// MI455X (gfx1250) — compile-verified
//
#include <hip/hip_runtime.h>

// ---------------------------------------------------------------------------
// InverseInDegreeEdgeWeighting for MI455X (gfx1250, wave32)
//
// out[e, :] = message[e, :] * (1 / bincount(target)[target[e]])
//
// Memory-bound (AI ~0.13 FLOP/B): WMMA not applicable. Strategy is pure
// bandwidth: b128 vector memory ops, L2-resident histogram via global
// atomics, non-temporal hints on the streaming message/out traffic.
// ---------------------------------------------------------------------------

typedef float v4f __attribute__((ext_vector_type(4)));
typedef int   v4i __attribute__((ext_vector_type(4)));

static constexpr int kNumNodes = 100000;   // fixed by reference (bincount length)
static constexpr int kDim      = 48;
static constexpr int kChunks   = kDim / 4; // 12 float4 chunks per edge row

// --- Kernel 1: clear the count array in workspace (every call) --------------
__global__ __launch_bounds__(256) void IIDEW_zero_counts(unsigned int* __restrict__ counts,
                                                         int n) {
    int i = blockIdx.x * 256 + threadIdx.x;
    if (i < n) counts[i] = 0u;
}

// --- Kernel 2: histogram of target with global atomics ----------------------
// counts fits in 400 KB -> fully L2 resident (192 MB L2); ~16 increments per
// bin so contention is negligible. Load 4 targets per thread via b128.
__global__ __launch_bounds__(256) void IIDEW_hist(const int* __restrict__ target,
                                                  unsigned int* __restrict__ counts,
                                                  int num_edges) {
    int tid    = blockIdx.x * 256 + threadIdx.x;
    int stride = gridDim.x * 256;
    int n4     = num_edges >> 2;

    const v4i* __restrict__ t4 = (const v4i*)target;
    for (int j = tid; j < n4; j += stride) {
        v4i t = t4[j];                       // global_load_b128
        atomicAdd(&counts[t.x], 1u);
        atomicAdd(&counts[t.y], 1u);
        atomicAdd(&counts[t.z], 1u);
        atomicAdd(&counts[t.w], 1u);
    }
    // tail (num_edges not multiple of 4)
    for (int e = (n4 << 2) + tid; e < num_edges; e += stride) {
        atomicAdd(&counts[target[e]], 1u);
    }
}

// --- Kernel 3: streaming scale ----------------------------------------------
// One thread per float4 chunk: consecutive lanes touch consecutive 16B chunks
// -> perfectly coalesced b128 traffic. target/counts loads are cached (hot in
// L2/WGP$ because 12 neighbouring lanes share the same edge); message/out use
// non-temporal hints since they are touched exactly once.
__global__ __launch_bounds__(256) void IIDEW_scale(const int* __restrict__ target,
                                                   const unsigned int* __restrict__ counts,
                                                   const v4f* __restrict__ msg,
                                                   v4f* __restrict__ out,
                                                   int total_chunks) {
    int idx = blockIdx.x * 256 + threadIdx.x;
    if (idx >= total_chunks) return;

    int e = idx / kChunks;                   // strength-reduced to mulhi magic
    int t = target[e];                       // cached load (shared by 12 lanes)
    unsigned int c = counts[t];              // L2-resident gather
    float w = 1.0f / (float)c;               // exact divide to match reference

    v4f m = __builtin_nontemporal_load(&msg[idx]);   // global_load_b128 (NT)
    v4f r;
    r.x = m.x * w;
    r.y = m.y * w;
    r.z = m.z * w;
    r.w = m.w * w;
    __builtin_nontemporal_store(r, &out[idx]);       // global_store_b128 (NT)
}

extern "C" void kernel_launch(void* const* d_in, const int* in_sizes, int n_in,
                              void* d_out, int out_size, void* d_ws, size_t ws_size,
                              hipStream_t stream) {
    // Inputs per reference setup_inputs(): 0=source (unused), 1=target, 2=message
    const int*   target  = (const int*)d_in[1];
    const float* message = (const float*)d_in[2];
    float*       out     = (float*)d_out;

    const int num_edges = in_sizes[1];

    // Workspace layout: counts[kNumNodes] (400 KB)
    unsigned int* counts = (unsigned int*)d_ws;

    // 1) zero counts (deterministic: redone every call)
    {
        int blocks = (kNumNodes + 255) / 256;
        IIDEW_zero_counts<<<blocks, 256, 0, stream>>>(counts, kNumNodes);
    }

    // 2) histogram
    {
        int n4 = num_edges >> 2;
        int blocks = (n4 + 255) / 256;
        if (blocks < 1) blocks = 1;
        if (blocks > 8192) blocks = 8192;    // grid-stride cap
        IIDEW_hist<<<blocks, 256, 0, stream>>>(target, counts, num_edges);
    }

    // 3) streaming scale: one thread per float4 (48 floats -> 12 chunks/edge)
    {
        int total_chunks = num_edges * kChunks;   // 19.2M, fits in int32
        int blocks = (total_chunks + 255) / 256;
        IIDEW_scale<<<blocks, 256, 0, stream>>>(target, counts,
                                                (const v4f*)message,
                                                (v4f*)out, total_chunks);
    }
}